// GradientGuidedSelectiveScan_36919538876646
// MI455X (gfx1250) — compile-verified
//
#include <hip/hip_runtime.h>
#include <math.h>

// Problem constants (match reference)
#define D_INNER 1024
#define LSEQ    2048
#define NSTATE  16
#define DTRANK  32
#define CROWS   64      // DTRANK + 2*NSTATE
#define BATCH   2

typedef __attribute__((ext_vector_type(2))) float v2f;
typedef __attribute__((ext_vector_type(8))) float v8f;

__device__ __forceinline__ float softplusf(float z) {
    // numerically stable softplus
    return (z > 20.0f) ? z : log1pf(__expf(z));
}

// ---------------------------------------------------------------------------
// Kernel 1: x_dbl[b,c,l] = sum_d W[c,d] * x[b,l,d]     (fp32 WMMA 16x16x4)
// Gradient-C term fused into rows c >= 48 at write-out.
// One wave computes one 16x16 output tile; K marches in steps of 4.
// A-tile lane layout (32-bit 16x4): lane<16 -> K={k0,k0+1}, lane>=16 -> K={k0+2,k0+3}
// => per lane a single contiguous float2 load for both A and B (K is the
//    contiguous 'd' axis of both W (row-major) and x (B,L,D)).
// ---------------------------------------------------------------------------
__global__ void __launch_bounds__(256)
xproj_wmma_kernel(const float* __restrict__ x,
                  const float* __restrict__ W,
                  const float* __restrict__ gs,
                  const float* __restrict__ gcw,
                  const float* __restrict__ gcb,
                  const float* __restrict__ gCw_p,
                  float* __restrict__ xdbl)
{
    const int lane = threadIdx.x & 31;
    const int wave = (blockIdx.x * blockDim.x + threadIdx.x) >> 5;
    // tiles: BATCH * (CROWS/16 = 4) * (LSEQ/16 = 128) = 1024
    const int b  = wave >> 9;
    const int r  = wave & 511;
    const int c0 = (r >> 7) << 4;
    const int l0 = (r & 127) << 4;

    const int row   = c0 + (lane & 15);
    const int col   = l0 + (lane & 15);
    const int khalf = (lane >> 4) << 1;          // 0 or 2

    const float* Arow = W + (size_t)row * D_INNER + khalf;
    const float* Bcol = x + ((size_t)b * LSEQ + col) * D_INNER + khalf;

    v8f acc = {};
    for (int k0 = 0; k0 < D_INNER; k0 += 4) {
        v2f a  = *(const v2f*)(Arow + k0);
        v2f bb = *(const v2f*)(Bcol + k0);
        acc = __builtin_amdgcn_wmma_f32_16x16x4_f32(false, a, false, bb,
                                                    (short)0, acc, false, false);
    }

    const float gCw   = gCw_p[0];
    const int   mbase = c0 + ((lane >> 4) << 3);
    #pragma unroll
    for (int v = 0; v < 8; ++v) {
        int   c   = mbase + v;
        float val = acc[v];
        if (c >= DTRANK + NSTATE) {
            // Cs row: replicate the reference's raw (B,L,N)->(B,N,L) reshape:
            // Cs[b,n,l] += gCw * (gs[b, n*128 + l/16]*gcw[l%16] + gcb[l%16])
            int n  = c - (DTRANK + NSTATE);
            int lp = n * (LSEQ / 16) + (col >> 4);
            int np = col & 15;
            val += gCw * (gs[b * LSEQ + lp] * gcw[np] + gcb[np]);
        }
        xdbl[((size_t)b * CROWS + c) * LSEQ + col] = val;
    }
}

// ---------------------------------------------------------------------------
// Kernel 2: delta[b,l,d] = softplus( sum_r dtw[d,r]*x_dbl[b,r,l] + dtb[d] )
// fp32 WMMA 16x16x4, K = DTRANK = 32 (8 steps). delta stored (B,L,D) so the
// scan kernel's per-timestep loads are coalesced across d.
// ---------------------------------------------------------------------------
__global__ void __launch_bounds__(256)
dtproj_wmma_kernel(const float* __restrict__ xdbl,
                   const float* __restrict__ dtw,
                   const float* __restrict__ dtb,
                   float* __restrict__ delta)
{
    const int lane = threadIdx.x & 31;
    const int wave = (blockIdx.x * blockDim.x + threadIdx.x) >> 5;
    // tiles: BATCH * (D_INNER/16 = 64) * (LSEQ/16 = 128) = 16384
    const int b  = wave >> 13;
    const int r  = wave & 8191;
    const int d0 = (r >> 7) << 4;
    const int l0 = (r & 127) << 4;

    const int drow  = d0 + (lane & 15);
    const int col   = l0 + (lane & 15);
    const int khalf = (lane >> 4) << 1;

    const float* Arow  = dtw + (size_t)drow * DTRANK + khalf;
    const float* Bbase = xdbl + (size_t)b * CROWS * LSEQ + col;

    v8f acc = {};
    #pragma unroll
    for (int k0 = 0; k0 < DTRANK; k0 += 4) {
        v2f a = *(const v2f*)(Arow + k0);
        v2f bb;
        bb.x = Bbase[(size_t)(k0 + khalf)     * LSEQ];   // B[k, l], K-stride = LSEQ
        bb.y = Bbase[(size_t)(k0 + khalf + 1) * LSEQ];
        acc = __builtin_amdgcn_wmma_f32_16x16x4_f32(false, a, false, bb,
                                                    (short)0, acc, false, false);
    }

    const int mbase = d0 + ((lane >> 4) << 3);
    #pragma unroll
    for (int v = 0; v < 8; ++v) {
        int d = mbase + v;
        delta[((size_t)b * LSEQ + col) * D_INNER + d] = softplusf(acc[v] + dtb[d]);
    }
}

// ---------------------------------------------------------------------------
// Kernel 3: selective scan. One thread per (b,d) row, h[16] in registers.
// B/C columns (shared by every d within a batch) staged through LDS in
// 16-timestep chunks -> one barrier pair per 16 steps.
// ---------------------------------------------------------------------------
__global__ void __launch_bounds__(256)
scan_kernel(const float* __restrict__ x,
            const float* __restrict__ xdbl,
            const float* __restrict__ delta,
            const float* __restrict__ A_logs,
            const float* __restrict__ Ds,
            float* __restrict__ out)
{
    __shared__ float sB[16][NSTATE];
    __shared__ float sC[16][NSTATE];

    const int t = blockIdx.x * blockDim.x + threadIdx.x;
    const int b = t / D_INNER;           // 256 | 1024 -> one batch per block
    const int d = t % D_INNER;

    float Arow[NSTATE];
    #pragma unroll
    for (int n = 0; n < NSTATE; ++n)
        Arow[n] = -__expf(A_logs[(size_t)d * NSTATE + n]);
    const float dsv = Ds[d];

    float h[NSTATE];
    #pragma unroll
    for (int n = 0; n < NSTATE; ++n) h[n] = 0.f;

    const int ll = threadIdx.x >> 4;     // which of 16 timesteps this thread stages
    const int nn = threadIdx.x & 15;     // which state component
    const float* Bsrc = xdbl + ((size_t)b * CROWS + DTRANK + nn) * LSEQ;
    const float* Csrc = Bsrc + (size_t)NSTATE * LSEQ;   // Cs (grad term already fused)

    for (int lb = 0; lb < LSEQ; lb += 16) {
        __syncthreads();
        sB[ll][nn] = Bsrc[lb + ll];
        sC[ll][nn] = Csrc[lb + ll];
        __syncthreads();

        for (int j = 0; j < 16; ++j) {
            const size_t base = ((size_t)b * LSEQ + (lb + j)) * D_INNER + d;
            const float dv = delta[base];
            const float xv = x[base];
            const float du = dv * xv;
            float y = 0.f;
            #pragma unroll
            for (int n = 0; n < NSTATE; ++n) {
                h[n] = __expf(dv * Arow[n]) * h[n] + du * sB[j][n];
                y   += h[n] * sC[j][n];
            }
            out[base] = y + xv * dsv;
        }
    }
}

extern "C" void kernel_launch(void* const* d_in, const int* in_sizes, int n_in,
                              void* d_out, int out_size, void* d_ws, size_t ws_size,
                              hipStream_t stream)
{
    const float* x    = (const float*)d_in[0];
    const float* gs   = (const float*)d_in[1];
    const float* Wxp  = (const float*)d_in[2];
    const float* dtw  = (const float*)d_in[3];
    const float* dtb  = (const float*)d_in[4];
    const float* Alog = (const float*)d_in[5];
    const float* Dsp  = (const float*)d_in[6];
    const float* gcw  = (const float*)d_in[7];
    const float* gcb  = (const float*)d_in[8];
    const float* gCw  = (const float*)d_in[9];
    float* out = (float*)d_out;

    float* xdbl  = (float*)d_ws;                                   // 1 MB
    float* delta = xdbl + (size_t)BATCH * CROWS * LSEQ;            // 16 MB

    // 1024 tile-waves -> 128 blocks of 8 waves
    xproj_wmma_kernel<<<128, 256, 0, stream>>>(x, Wxp, gs, gcw, gcb, gCw, xdbl);
    // 16384 tile-waves -> 2048 blocks
    dtproj_wmma_kernel<<<2048, 256, 0, stream>>>(xdbl, dtw, dtb, delta);
    // 2048 scan rows -> 8 blocks
    scan_kernel<<<8, 256, 0, stream>>>(x, xdbl, delta, Alog, Dsp, out);
}